// Mapping_32590211842713
// MI455X (gfx1250) — compile-verified
//
#include <hip/hip_runtime.h>
#include <hip/hip_bf16.h>

#define RES    64
#define N_PHAL 20

typedef __attribute__((ext_vector_type(2))) float v2f;
typedef __attribute__((ext_vector_type(8))) float v8f;

// ---------------------------------------------------------------------------
// Kernel 1: one 256-thread block per (b, phalanx) plane.
// Each thread: 4 groups of 4 pixels (5x b128 loads per group, 16B aligned).
// Accumulate den, vfx, vfy, vsx, vsy in registers; stage 5x256 partials in
// LDS; wave 0 reduces each quantity with chained V_WMMA_F32_16X16X4_F32
// against an all-ones B matrix (layout-independent sum), then shfl-xor to
// merge the two row halves.
// ---------------------------------------------------------------------------
__global__ __launch_bounds__(256) void phal_reduce_kernel(
    const float* __restrict__ x, float* __restrict__ ws) {
  const int plane = blockIdx.x;                       // b * N_PHAL + p
  const float* base = x + (size_t)plane * (RES * RES * 5);
  const int t = threadIdx.x;                          // 0..255

  float den = 0.f, vfx = 0.f, vfy = 0.f, vsx = 0.f, vsy = 0.f;

#pragma unroll
  for (int k = 0; k < 4; ++k) {
    const int g = t + k * 256;                        // 4-pixel group index
    const float4* p4 = (const float4*)(base + g * 20);
    float4 v0 = p4[0], v1 = p4[1], v2 = p4[2], v3 = p4[3], v4 = p4[4];
    const float px[4][5] = {
        {v0.x, v0.y, v0.z, v0.w, v1.x},
        {v1.y, v1.z, v1.w, v2.x, v2.y},
        {v2.z, v2.w, v3.x, v3.y, v3.z},
        {v3.w, v4.x, v4.y, v4.z, v4.w}};
    const int pix0 = g * 4;
#pragma unroll
    for (int u = 0; u < 4; ++u) {
      const int pix = pix0 + u;
      const float fi = (float)(pix >> 6);             // locs[...,0] = i
      const float fj = (float)(pix & 63);             // locs[...,1] = j
      const float fx = px[u][0], fy = px[u][1], m = px[u][2];
      const float d0 = px[u][3], d1 = px[u][4];
      const float am   = fabsf(m);
      const float dirx = fy * m;
      const float diry = -fx * m;
      den += am;
      vfx += am * ( dirx * d0 + fi);
      vfy += am * ( diry * d0 + fj);
      vsx += am * (-dirx * d1 + fi);
      vsy += am * (-diry * d1 + fj);
    }
  }

  __shared__ float red[5][256];
  red[0][t] = den;
  red[1][t] = vfx;
  red[2][t] = vfy;
  red[3][t] = vsx;
  red[4][t] = vsy;
  __syncthreads();

  if (t < 32) {                                       // wave 0, EXEC all ones
    v2f ones;
    ones[0] = 1.0f;
    ones[1] = 1.0f;
    float out5[5];
#pragma unroll
    for (int q = 0; q < 5; ++q) {
      v8f d = {};
#pragma unroll
      for (int it = 0; it < 4; ++it) {
        v2f a;
        a[0] = red[q][it * 64 + t];
        a[1] = red[q][it * 64 + 32 + t];
        // D[m,n] = sum_k A[m,k] * 1.0 + C[m,n]  (layout-independent w/ B=1)
        d = __builtin_amdgcn_wmma_f32_16x16x4_f32(
            /*neg_a=*/false, a, /*neg_b=*/false, ones,
            /*c_mod=*/(short)0, d, /*reuse_a=*/false, /*reuse_b=*/false);
      }
      // lanes 0-15 hold rows 0..7, lanes 16-31 rows 8..15 -> fold 8 VGPRs,
      // then merge the two halves across the wave.
      float s = d[0] + d[1] + d[2] + d[3] + d[4] + d[5] + d[6] + d[7];
      s += __shfl_xor(s, 16, 32);
      out5[q] = s;
    }
    if (t == 0) {
      float* w = ws + (size_t)plane * 5;
#pragma unroll
      for (int q = 0; q < 5; ++q) w[q] = out5[q];
    }
  }
}

// ---------------------------------------------------------------------------
// Kernel 2: one block per batch; finalize 21x2 keypoints from 20 centroids.
// ---------------------------------------------------------------------------
__global__ __launch_bounds__(64) void finalize_kernel(
    const float* __restrict__ ws, float* __restrict__ out) {
  const int b = blockIdx.x;
  const int t = threadIdx.x;                          // 0..63
  const float* w = ws + (size_t)b * N_PHAL * 5;

  __shared__ float denS[N_PHAL];
  __shared__ float vfS[N_PHAL][2];
  __shared__ float vsS[N_PHAL][2];

  if (t < N_PHAL) {
    const float d  = w[t * 5 + 0];
    const float ds = (d == 0.0f) ? 1.0f : d;
    denS[t]   = d;
    vfS[t][0] = w[t * 5 + 1] / ds;
    vfS[t][1] = w[t * 5 + 2] / ds;
    vsS[t][0] = w[t * 5 + 3] / ds;
    vsS[t][1] = w[t * 5 + 4] / ds;
  }
  __syncthreads();

  if (t < 42) {
    const int q = t >> 1;                             // output row 0..20
    const int c = t & 1;
    float r;
    if (q == 0) {
      float s = 0.f;
#pragma unroll
      for (int rt = 0; rt < N_PHAL; rt += 4)
        s += (denS[rt] != 0.0f) ? vfS[rt][c] : 0.0f;
      r = s * 0.2f;                                   // / N_FINGERS
    } else {
      const int  iq  = (q - 1) >> 2;
      const int  jq  = 4 * iq + 4 - q;
      const int  pq  = 4 * iq + jq;
      const int  pn  = min(pq + 1, N_PHAL - 1);
      const bool tip = (jq == 3);
      r = tip ? vsS[pq][c] : (vsS[pq][c] + vfS[pn][c]) * 0.5f;
    }
    out[(size_t)b * 42 + t] = r;
  }
}

extern "C" void kernel_launch(void* const* d_in, const int* in_sizes, int n_in,
                              void* d_out, int out_size, void* d_ws, size_t ws_size,
                              hipStream_t stream) {
  const float* x  = (const float*)d_in[0];
  float* out      = (float*)d_out;
  float* ws       = (float*)d_ws;    // needs B*N_PHAL*5 floats = 100 KiB
  const int B     = in_sizes[0] / (N_PHAL * RES * RES * 5);   // 256

  phal_reduce_kernel<<<B * N_PHAL, 256, 0, stream>>>(x, ws);
  finalize_kernel<<<B, 64, 0, stream>>>(ws, out);
}